// minDistLoss_75711683493951
// MI455X (gfx1250) — compile-verified
//
#include <hip/hip_runtime.h>
#include <hip/hip_bf16.h>
#include <stdint.h>

// Problem geometry (fixed by the reference).
#define NB    64                 // batch
#define SS    16                 // samples per batch
#define CHW   49152              // 3*128*128 floats per image
#define TPB   256                // threads per block (8 wave32s)
#define NCH   (CHW / 4 / TPB)    // 48 chunks of 256 float4 per (n,s) pair
#define DEPTH 4                  // async-to-LDS pipeline depth
#define CHUNK_BYTES (TPB * 16)   // 4096 B per chunk

typedef __attribute__((address_space(3))) float lds_f32;

__global__ __launch_bounds__(TPB) void minDist_partial(
    const float* __restrict__ inputs,
    const float* __restrict__ samples,
    float* __restrict__ pair_loss)
{
    __shared__ float4 stage[DEPTH][TPB];   // 16 KB ring buffer
    __shared__ float  red[TPB / 32];

    const int tid = threadIdx.x;
    const int p   = blockIdx.x;            // pair id: n = p/SS, s = p%SS
    const int n   = p / SS;

    const float4* __restrict__ inp4 =
        (const float4*)(inputs + (size_t)n * CHW);

    // Uniform 64-bit base of this pair's sample row (GVS saddr),
    // per-lane 32-bit byte offset (vaddr).
    const unsigned long long sbase =
        (unsigned long long)(uintptr_t)(samples + (size_t)p * CHW);
    const unsigned voff = (unsigned)tid * 16u;

    // Issue one async B128 load per lane: copies 16 bytes from global address gb + voff to LDS address laddr.
    auto prefetch = [&](int c) {
        lds_f32* laddr =
            (lds_f32*)&stage[c & (DEPTH - 1)][tid];   // 32-bit as3 address
        unsigned long long gb =
            sbase + (unsigned long long)c * CHUNK_BYTES;
        asm volatile("global_load_async_to_lds_b128 %0, %1, %2"
                     :: "v"(laddr), "v"(voff), "s"(gb)
                     : "memory");
    };

    // Fill the pipeline.
    #pragma unroll
    for (int c = 0; c < DEPTH; ++c) prefetch(c);

    float4 acc = make_float4(0.f, 0.f, 0.f, 0.f);

    for (int c = 0; c < NCH; ++c) {
        // Retire chunk c: allow min(DEPTH-1, chunks-still-inflight) outstanding.
        int allow = NCH - 1 - c;
        if (allow > DEPTH - 1) allow = DEPTH - 1;
        switch (allow) {
            case 3:  asm volatile("s_wait_asynccnt 0x3" ::: "memory"); break;
            case 2:  asm volatile("s_wait_asynccnt 0x2" ::: "memory"); break;
            case 1:  asm volatile("s_wait_asynccnt 0x1" ::: "memory"); break;
            default: asm volatile("s_wait_asynccnt 0x0" ::: "memory"); break;
        }

        const int buf = c & (DEPTH - 1);
        float4 sv = stage[buf][tid];               // ds_load_b128
        float4 iv = inp4[c * TPB + tid];           // global_load_b128 (L2-hot)

        float dx = sv.x - iv.x;
        float dy = sv.y - iv.y;
        float dz = sv.z - iv.z;
        float dw = sv.w - iv.w;
        acc.x = fmaf(dx, dx, acc.x);
        acc.y = fmaf(dy, dy, acc.y);
        acc.z = fmaf(dz, dz, acc.z);
        acc.w = fmaf(dw, dw, acc.w);

        if (c + DEPTH < NCH) {
            // WAR guard: our ds_load of this ring slot must land before the
            // async engine overwrites it.
            asm volatile("s_wait_dscnt 0x0" ::: "memory");
            prefetch(c + DEPTH);
        }
    }

    // Block reduction: horizontal add, wave32 shuffle tree, LDS across 8 waves.
    float a = (acc.x + acc.y) + (acc.z + acc.w);
    #pragma unroll
    for (int m = 16; m > 0; m >>= 1) a += __shfl_xor(a, m, 32);
    if ((tid & 31) == 0) red[tid >> 5] = a;
    __syncthreads();
    if (tid < (TPB / 32)) {
        float r = red[tid];
        #pragma unroll
        for (int m = (TPB / 64); m > 0; m >>= 1) r += __shfl_xor(r, m, 32);
        if (tid == 0) pair_loss[p] = r;
    }
}

// min over S per batch, sum over N -> scalar.
__global__ __launch_bounds__(64) void minDist_final(
    const float* __restrict__ pair_loss, float* __restrict__ out)
{
    __shared__ float red[2];
    const int t = threadIdx.x;          // 0..63 == batch index
    float m = pair_loss[t * SS];
    #pragma unroll
    for (int s = 1; s < SS; ++s) m = fminf(m, pair_loss[t * SS + s]);
    #pragma unroll
    for (int k = 16; k > 0; k >>= 1) m += __shfl_xor(m, k, 32);
    if ((t & 31) == 0) red[t >> 5] = m;
    __syncthreads();
    if (t == 0) out[0] = red[0] + red[1];
}

extern "C" void kernel_launch(void* const* d_in, const int* in_sizes, int n_in,
                              void* d_out, int out_size, void* d_ws, size_t ws_size,
                              hipStream_t stream) {
    const float* inputs  = (const float*)d_in[0];   // [64,3,128,128] f32
    const float* samples = (const float*)d_in[1];   // [64,16,3,128,128] f32
    float* pair_loss = (float*)d_ws;                // 1024 floats scratch
    float* out = (float*)d_out;                     // scalar f32

    minDist_partial<<<NB * SS, TPB, 0, stream>>>(inputs, samples, pair_loss);
    minDist_final<<<1, 64, 0, stream>>>(pair_loss, out);
}